// PointPillarBaselineLate_9629316678049
// MI455X (gfx1250) — compile-verified
//
#include <hip/hip_runtime.h>
#include <hip/hip_bf16.h>

// Problem constants (from reference)
#define HH   200
#define WW   704
#define HWP  140800      // H*W pixels
#define CIN  256
#define AA   2
#define MROW 16          // 2 cls + 14 reg output channels
#define NCAV 4
#define KSEL 20000
#define FLAT 281600      // A*H*W

typedef __attribute__((ext_vector_type(2))) float        v2f;
typedef __attribute__((ext_vector_type(8))) float        v8f;
typedef __attribute__((ext_vector_type(4))) unsigned int v4u;
typedef __attribute__((ext_vector_type(8))) int          v8i;
typedef __attribute__((ext_vector_type(4))) int          v4i;

__device__ __forceinline__ float sigmoidf(float x) {
    return 1.0f / (1.0f + __expf(-x));
}

__global__ __launch_bounds__(256)
void fused_head_late_fusion(const float* __restrict__ feat,      // [256, H, W]
                            const float* __restrict__ cls_w,     // [2, 256]
                            const float* __restrict__ cls_b,     // [2]
                            const float* __restrict__ reg_w,     // [14, 256]
                            const float* __restrict__ reg_b,     // [14]
                            const float* __restrict__ psm_v2x,   // [4, 20000]
                            const float* __restrict__ rm_v2x,    // [4, 7*20000]
                            const float* __restrict__ scores_v2x,// [4, 20000]
                            const int*   __restrict__ mask_idx,  // [4, 20000] sorted
                            float* __restrict__ out_psm,         // [2, H, W]
                            float* __restrict__ out_rm)          // [14, H, W]
{
    // LDS: per-wave B tiles (TDM destination, [k][16] layout) 128 KB,
    //      weights transposed [k][m] 16 KB, per-wave D tiles 8 KB.
    __shared__ float fsh[8][CIN][16];
    __shared__ float wts[CIN][MROW];
    __shared__ float dsh[8][MROW][16];
    __shared__ float biassh[MROW];

    const int tid  = threadIdx.x;
    const int warp = __builtin_amdgcn_readfirstlane(tid >> 5);  // wave-uniform
    const int lane = tid & 31;
    const int half = lane >> 4;   // 0: rows 0-7 / anchor 0 ; 1: rows 8-15 / anchor 1
    const int nn   = lane & 15;   // pixel-in-tile (also M for A-fragment reads)

    // ---- one wave == one 16-pixel tile ----
    const int tile = blockIdx.x * 8 + warp;          // 8800 tiles total, exact
    const int base = tile * 16;                      // first pixel of tile

    // ---- TDM: DMA the whole 16x256 B tile (16 KB) into this wave's LDS slab ----
    // D# group 0: count=1 | lds_addr | global_addr(57b) | type=2
    unsigned long long gaddr =
        (unsigned long long)(uintptr_t)(feat) + (unsigned long long)base * 4ull;
    unsigned lds_off = (unsigned)(uintptr_t)(&fsh[warp][0][0]);  // flat low 32b = LDS offset
    v4u g0;
    g0.x = 1u;                                            // count=1 (valid descriptor)
    g0.y = lds_off;                                       // lds_addr (bytes)
    g0.z = (unsigned)(gaddr & 0xffffffffull);             // global_addr[31:0]
    g0.w = (unsigned)((gaddr >> 32) & 0x01ffffffull)      // global_addr[56:32]
         | 0x80000000u;                                   // type=2 ("image") at [127:126]
    // D# group 1: data_size=4B; tensor = [256 rows][140800 cols]; tile = 16 x 256
    v8i g1;
    g1[0] = 0x20000;                                      // data_size=2 (4 bytes)
    g1[1] = (int)((HWP & 0xffff) << 16);                  // tensor_dim0[15:0] at bits 63:48
    g1[2] = (int)(((unsigned)HWP >> 16) | (CIN << 16));   // tensor_dim0[31:16] | tensor_dim1[15:0]
    g1[3] = (int)(16 << 16);                              // tile_dim0 = 16
    g1[4] = CIN;                                          // tile_dim1 = 256 (tile_dim2 = 0)
    g1[5] = HWP;                                          // tensor_dim0_stride[31:0]
    g1[6] = 0;                                            // stride[47:32] | dim1_stride[15:0]
    g1[7] = 0;
    v4i g2 = {0, 0, 0, 0};                                // 2-D tensor: groups 2/3 unused
    v4i g3 = {0, 0, 0, 0};
    v8i g4 = {0, 0, 0, 0, 0, 0, 0, 0};                    // extra group (clang-23 6-arg form)
    __builtin_amdgcn_tensor_load_to_lds(g0, g1, g2, g3, g4, /*cpol=*/0);

    // ---- overlap: cooperative weight stage wts[k][m] while TDM streams ----
    for (int t = tid; t < CIN * MROW; t += 256) {
        int k = t >> 4;
        int m = t & 15;
        wts[k][m] = (m < AA) ? cls_w[m * CIN + k] : reg_w[(m - AA) * CIN + k];
    }
    if (tid < MROW) {
        biassh[tid] = (tid < AA) ? cls_b[tid] : reg_b[tid - AA];
    }
    __syncthreads();

    // ---- wait for this wave's tensor DMA, then GEMM entirely out of LDS ----
    __builtin_amdgcn_s_wait_tensorcnt(0);
    asm volatile("" ::: "memory");   // compiler: TDM wrote fsh; don't reorder/fold ds loads

    v8f c = {};   // 16x16 f32 accumulator
    #pragma unroll 4
    for (int k0 = 0; k0 < CIN; k0 += 4) {
        v2f a, b;
        // A (16x4): lane m = nn, K = k0 + 2*half + v
        a.x = wts[k0 + 2 * half + 0][nn];
        a.y = wts[k0 + 2 * half + 1][nn];
        // B (4x16): K = k0 + 2*half + v, N = nn  (from TDM-staged LDS tile)
        b.x = fsh[warp][k0 + 2 * half + 0][nn];
        b.y = fsh[warp][k0 + 2 * half + 1][nn];
        c = __builtin_amdgcn_wmma_f32_16x16x4_f32(
                /*neg_a=*/false, a, /*neg_b=*/false, b,
                /*c_mod=*/(short)0, c, /*reuse_a=*/false, /*reuse_b=*/false);
    }

    // ---- stage D (+bias) to LDS so each lane can read arbitrary channels ----
    #pragma unroll
    for (int v = 0; v < 8; ++v) {
        int m = v + 8 * half;
        dsh[warp][m][nn] = c[v] + biassh[m];
    }
    asm volatile("" ::: "memory");   // wave-local LDS RAW; backend inserts s_wait_dscnt

    // ---- fused restore + argmax + winner gather ----
    // lane handles anchor a = half, pixel P = base + nn, flat cell j = a*HW + P
    const int a = half;
    const int P = base + nn;
    const int j = a * HWP + P;

    float ego_psm = dsh[warp][a][nn];
    float best    = sigmoidf(ego_psm);
    int   wc      = -1;              // -1 == ego wins
    int   wk      = 0;

    #pragma unroll
    for (int i = 0; i < NCAV; ++i) {
        const int* mi = mask_idx + i * KSEL;
        int lo = 0, hi = KSEL - 1, found = -1;
        while (lo <= hi) {
            int mid = (lo + hi) >> 1;
            int vv  = mi[mid];
            if (vv == j) { found = mid; break; }
            else if (vv < j) lo = mid + 1;
            else hi = mid - 1;
        }
        if (found >= 0) {
            float s = scores_v2x[i * KSEL + found];
            if (s > best) { best = s; wc = i; wk = found; }  // strict: argmax tie -> first
        }
    }

    // psm output
    out_psm[j] = (wc < 0) ? ego_psm : psm_v2x[wc * KSEL + wk];

    // rm output: reg channel rr = 2r + a uses the same winner as anchor a
    #pragma unroll
    for (int r = 0; r < 7; ++r) {
        int rr = 2 * r + a;
        float val = (wc < 0) ? dsh[warp][rr + 2][nn]
                             : rm_v2x[(size_t)wc * (7 * KSEL) + r * KSEL + wk];
        out_rm[(size_t)rr * HWP + P] = val;
    }
}

extern "C" void kernel_launch(void* const* d_in, const int* in_sizes, int n_in,
                              void* d_out, int out_size, void* d_ws, size_t ws_size,
                              hipStream_t stream) {
    const float* feat       = (const float*)d_in[0];
    const float* cls_w      = (const float*)d_in[1];
    const float* cls_b      = (const float*)d_in[2];
    const float* reg_w      = (const float*)d_in[3];
    const float* reg_b      = (const float*)d_in[4];
    const float* psm_v2x    = (const float*)d_in[5];
    const float* rm_v2x     = (const float*)d_in[6];
    const float* scores_v2x = (const float*)d_in[7];
    const int*   mask_idx   = (const int*)d_in[8];
    // d_in[9] (mask_reg_idx) is derivable: offsets r*FLAT — not needed.

    float* out_psm = (float*)d_out;          // [2, 200, 704]  = 281600
    float* out_rm  = out_psm + FLAT;         // [14, 200, 704] = 1971200

    // 8800 tiles of 16 pixels, 8 waves (tiles) per 256-thread block
    fused_head_late_fusion<<<1100, 256, 0, stream>>>(
        feat, cls_w, cls_b, reg_w, reg_b,
        psm_v2x, rm_v2x, scores_v2x, mask_idx,
        out_psm, out_rm);
}